// GatingNetwork_13116830122384
// MI455X (gfx1250) — compile-verified
//
#include <hip/hip_runtime.h>
#include <math.h>

typedef __attribute__((ext_vector_type(2))) float v2f;
typedef __attribute__((ext_vector_type(4))) float v4f;
typedef __attribute__((ext_vector_type(8))) float v8f;
typedef __attribute__((ext_vector_type(4))) int   v4i;

#define DDIM 2048
#define NEXP 64
#define TM   128          // tokens per block
#define TN   128          // 64 gate cols + 64 noise cols fused
#define KT   32           // K tile
#define NKT  (DDIM / KT)  // 64 K tiles
#define LDA  36           // row stride (floats): 16B-aligned rows, conflict-free b64 frag reads
#define CST  129          // C scratch stride
#define BUFOFF ((TM + TN) * LDA)  // floats per staging buffer
#define NOISE_EPS 0.01f
#define INV_SQRT2 0.70710678118654752f

#if __has_builtin(__builtin_amdgcn_global_load_async_to_lds_b128)
#define HAS_ASYNC 1
#else
#define HAS_ASYNC 0
#endif

#if HAS_ASYNC && __has_builtin(__builtin_amdgcn_s_wait_asynccnt)
#define WAIT_ASYNC(n) __builtin_amdgcn_s_wait_asynccnt(n)
#elif HAS_ASYNC
#define WAIT_ASYNC(n) asm volatile("s_wait_asynccnt " #n ::: "memory")
#else
#define WAIT_ASYNC(n) ((void)0)
#endif

typedef __attribute__((address_space(1))) v4i  g_v4i;
typedef __attribute__((address_space(3))) v4i  l_v4i;

__device__ __forceinline__ float softplus_f(float v) {
    return (v > 20.0f) ? v : log1pf(expf(v));
}

// copy 16 contiguous bytes global -> LDS (async on CDNA5, sync fallback)
__device__ __forceinline__ void stage16(const float* __restrict__ g, float* l) {
#if HAS_ASYNC
    __builtin_amdgcn_global_load_async_to_lds_b128(
        (g_v4i*)(void*)g,
        (l_v4i*)(void*)l,
        0, 0);
#else
    *(v4f*)l = *(const v4f*)g;
#endif
}

// stage one K-tile: A rows (x) + fused B rows (w_gate | w_noise), row-major, stride LDA
// 8 x b128 per thread => 8 ASYNCcnt increments per wave per tile
__device__ __forceinline__ void stage_tile(
    const float* __restrict__ x,
    const float* __restrict__ w_gate,
    const float* __restrict__ w_noise,
    float* As, float* Bs, int rowM0, int k0, int tid)
{
    #pragma unroll
    for (int it = 0; it < 4; ++it) {
        int flat = tid + it * 256;          // 1024 float4 chunks: 128 rows x 8
        int m  = flat >> 3;
        int kq = (flat & 7) << 2;
        stage16(x + (size_t)(rowM0 + m) * DDIM + k0 + kq, As + m * LDA + kq);
    }
    #pragma unroll
    for (int it = 0; it < 4; ++it) {
        int flat = tid + it * 256;
        int c  = flat >> 3;
        int kq = (flat & 7) << 2;
        const float* src = (c < NEXP) ? (w_gate + (size_t)c * DDIM)
                                      : (w_noise + (size_t)(c - NEXP) * DDIM);
        stage16(src + k0 + kq, Bs + c * LDA + kq);
    }
}

__global__ __launch_bounds__(256) void moe_router_kernel(
    const float* __restrict__ x,        // [N, D]
    const float* __restrict__ w_gate,   // [E, D]
    const float* __restrict__ w_noise,  // [E, D]
    const float* __restrict__ noise,    // [N, E]
    float* __restrict__ out_rw,         // [N, 2]
    int*   __restrict__ out_sel,        // [N, 2]
    float* __restrict__ out_load)       // [N, E]
{
    // double-buffered staging: 2 * (TM+TN) * LDA = 18432 floats (73.7 KB)
    // reused as C scratch (TM * CST = 16512 floats) after the GEMM loop
    __shared__ __align__(16) float lds[2 * BUFOFF];
    float* Cs = lds;

    const int tid   = threadIdx.x;
    const int wave  = tid >> 5;        // 0..7 -> M-tile
    const int lane  = tid & 31;
    const int grp   = lane >> 4;       // 0: K0/K1 ; 1: K2/K3
    const int ln16  = lane & 15;
    const int rowM0 = blockIdx.x * TM;
    const int mrow  = wave * 16 + ln16;

    v8f acc[8] = {};                   // wave's 16 rows x 128 cols

    // prologue: tile 0 in flight
    stage_tile(x, w_gate, w_noise, lds, lds + TM * LDA, rowM0, 0, tid);

    for (int t = 0; t < NKT; ++t) {
        float* cur = lds + (t & 1) * BUFOFF;
        float* nxt = lds + ((t + 1) & 1) * BUFOFF;
        if (t + 1 < NKT) {
            stage_tile(x, w_gate, w_noise, nxt, nxt + TM * LDA,
                       rowM0, (t + 1) * KT, tid);
            WAIT_ASYNC(8);             // tile t's 8 async ops done (in-order)
        } else {
            WAIT_ASYNC(0);
        }
        __syncthreads();               // all waves' copies for tile t landed

        const float* As = cur;
        const float* Bs = cur + TM * LDA;
        #pragma unroll
        for (int kk = 0; kk < KT; kk += 4) {
            // A frag: 2 consecutive K values, one aligned b64 LDS read
            v2f a = *(const v2f*)(As + mrow * LDA + kk + grp * 2);
            #pragma unroll
            for (int nt = 0; nt < 8; ++nt) {
                v2f b = *(const v2f*)(Bs + (nt * 16 + ln16) * LDA + kk + grp * 2);
                acc[nt] = __builtin_amdgcn_wmma_f32_16x16x4_f32(
                    false, a, false, b, (short)0, acc[nt], false, false);
            }
        }
        __syncthreads();               // done reading cur before it's re-staged
    }

    // ---- spill accumulators to LDS (C layout: row = token-in-block, col 0..127)
    #pragma unroll
    for (int nt = 0; nt < 8; ++nt) {
        #pragma unroll
        for (int r = 0; r < 8; ++r) {
            int row = wave * 16 + grp * 8 + r;
            int col = nt * 16 + ln16;
            Cs[row * CST + col] = acc[nt][r];
        }
    }
    __syncthreads();

    // ---- per-token epilogue: threads 0..127 own one token each
    if (tid < TM) {
        const int m = tid;
        const int n = rowM0 + m;
        const float* crow = Cs + m * CST;
        const float* nrow = noise + (size_t)n * NEXP;

        // pass 1: top-3 of noisy logits (strict > keeps earliest index on ties)
        float v0 = -INFINITY, v1 = -INFINITY, v2 = -INFINITY;
        int i0 = 0, i1 = 0;
        for (int e = 0; e < NEXP; ++e) {
            float g  = crow[e];
            float sd = softplus_f(crow[NEXP + e]) + NOISE_EPS;
            float s  = fmaf(nrow[e], sd, g);
            if (s > v0)      { v2 = v1; v1 = v0; i1 = i0; v0 = s; i0 = e; }
            else if (s > v1) { v2 = v1; v1 = s;  i1 = e; }
            else if (s > v2) { v2 = s; }
        }
        // softmax over top-2 (renormalized softmax == this)
        float p1  = expf(v1 - v0);
        float inv = 1.0f / (1.0f + p1);
        out_rw[n * 2 + 0]  = inv;
        out_rw[n * 2 + 1]  = p1 * inv;
        out_sel[n * 2 + 0] = i0;
        out_sel[n * 2 + 1] = i1;

        // pass 2: load = Phi((g - thr)/sd), thr = v2 if would-be-in else v1
        for (int e = 0; e < NEXP; ++e) {
            float g  = crow[e];
            float sd = softplus_f(crow[NEXP + e]) + NOISE_EPS;
            float s  = fmaf(nrow[e], sd, g);
            float z  = (s > v2) ? (g - v2) : (g - v1);
            out_load[(size_t)n * NEXP + e] =
                0.5f * (1.0f + erff((z / sd) * INV_SQRT2));
        }
    }
}

// expert_mask[e][j][n] = (selected_experts[n][j] == e), fully coalesced writes
__global__ __launch_bounds__(256) void expert_mask_kernel(
    const int* __restrict__ sel, int* __restrict__ mask, int ntok)
{
    int idx = blockIdx.x * blockDim.x + threadIdx.x;
    int total = NEXP * 2 * ntok;
    if (idx >= total) return;
    int n = idx % ntok;
    int j = (idx / ntok) & 1;
    int e = idx / (2 * ntok);
    mask[idx] = (sel[n * 2 + j] == e) ? 1 : 0;
}

extern "C" void kernel_launch(void* const* d_in, const int* in_sizes, int n_in,
                              void* d_out, int out_size, void* d_ws, size_t ws_size,
                              hipStream_t stream) {
    const float* x       = (const float*)d_in[0];
    const float* w_gate  = (const float*)d_in[1];
    const float* w_noise = (const float*)d_in[2];
    const float* noise   = (const float*)d_in[3];

    const int ntok = in_sizes[0] / DDIM;   // 16384

    // output layout (flat concat in return order):
    //   routing_weights f32 [N,2] | selected_experts i32 [N,2]
    //   expert_mask i32 [E,2,N]   | load f32 [N,E]
    float* out       = (float*)d_out;
    float* out_rw    = out;
    int*   out_sel   = (int*)(out + 2 * (size_t)ntok);
    int*   out_mask  = (int*)(out + 4 * (size_t)ntok);
    float* out_load  = out + 4 * (size_t)ntok + (size_t)NEXP * 2 * ntok;

    moe_router_kernel<<<ntok / TM, 256, 0, stream>>>(
        x, w_gate, w_noise, noise, out_rw, out_sel, out_load);

    int mask_total = NEXP * 2 * ntok;
    expert_mask_kernel<<<(mask_total + 255) / 256, 256, 0, stream>>>(
        out_sel, out_mask, ntok);
}